// MyGAT_1254130450647
// MI455X (gfx1250) — compile-verified
//
#include <hip/hip_runtime.h>

// ---------------------------------------------------------------------------
// MyGAT on MI455X (gfx1250).
//
// Key identity: att = softmax(attx, axis=0)  =>  every column of att sums to 1,
// so  out[y] = sum_x node[y]*att[x,y] = node[y].  The whole W_pair/a_pair/adj
// attention pipeline cancels algebraically.
//
//   out[c] = sum_{f=0}^{63} W_out[f*1024 + c] * h[f*1024 + c],  c = i*32+j
//
// Roofline: faithful path streams 268 MB of W_pair (~11.5 us at 23.3 TB/s,
// pure bandwidth); collapsed path reads 512 KB (~22 ns) -> launch-bound.
//
// The 1024 length-64 dot products are executed on the CDNA5 matrix pipe with
// full-precision V_WMMA_F32_16X16X4_F32 (no f16 rounding of inputs): each wave
// owns 16 cells; A = 16 W_out rows (16x64 over 16 chained K=4 steps),
// B = the matching 16 h columns (64x16); results are the diagonal of the
// 16x16 fp32 accumulator.
// ---------------------------------------------------------------------------

typedef __attribute__((ext_vector_type(2))) float v2f;
typedef __attribute__((ext_vector_type(8))) float v8f;

#define GAT_F      64
#define GAT_NCELL  1024

#if defined(__has_builtin)
#  if __has_builtin(__builtin_amdgcn_wmma_f32_16x16x4_f32)
#    define HAVE_WMMA_F32_16X16X4 1
#  endif
#endif

__global__ __launch_bounds__(256) void gat_collapsed_wmma(
    const float* __restrict__ H,   // h:     [64][1024] flat (f*1024 + i*32 + j)
    const float* __restrict__ W,   // W_out: [64][1024] flat (same layout)
    float* __restrict__ out)       // out:   [1024]     (i*32 + j)
{
    const int lane = threadIdx.x & 31;
    const int wid  = blockIdx.x * (blockDim.x >> 5) + (threadIdx.x >> 5);
    const int c0   = wid * 16;                 // 16 cells per wave, 64 waves total

#if defined(HAVE_WMMA_F32_16X16X4)
    // A 16x4 fp32 layout: lanes 0-15 -> M=lane,    K = {4t+0, 4t+1}
    //                     lanes16-31 -> M=lane-16, K = {4t+2, 4t+3}
    // B 4x16 fp32 layout mirrors A (row striped across lanes).
    const int m    = lane & 15;                // row (A) == col (B) index
    const int koff = (lane >> 4) << 1;         // 0 for low half-wave, 2 for high

    v8f acc = {};
#pragma unroll
    for (int t = 0; t < 16; ++t) {
        const int f0 = 4 * t + koff;
        v2f a, b;
        a.x = W[(f0    ) * GAT_NCELL + c0 + m];
        a.y = W[(f0 + 1) * GAT_NCELL + c0 + m];
        b.x = H[(f0    ) * GAT_NCELL + c0 + m];
        b.y = H[(f0 + 1) * GAT_NCELL + c0 + m];
        // 8 args: (neg_a, A, neg_b, B, c_mod, C, reuse_a, reuse_b)
        acc = __builtin_amdgcn_wmma_f32_16x16x4_f32(
            false, a, false, b, (short)0, acc, false, false);
    }

    // D 16x16 fp32: VGPR v, lanes 0-15 -> M=v, lanes 16-31 -> M=v+8; N=lane&15.
    // Diagonal (M==N) owners: lanes 0-7 (component = lane) and
    //                         lanes 24-31 (component = lane-24).
    const int  r     = lane & 15;                        // output row within block
    const bool owner = (lane < 8) || (lane >= 24);
    const int  v     = (lane < 8) ? lane : (lane - 24);  // accumulator component
    float d = acc[0];
    d = (v == 1) ? acc[1] : d;
    d = (v == 2) ? acc[2] : d;
    d = (v == 3) ? acc[3] : d;
    d = (v == 4) ? acc[4] : d;
    d = (v == 5) ? acc[5] : d;
    d = (v == 6) ? acc[6] : d;
    d = (v == 7) ? acc[7] : d;
    if (owner) out[c0 + r] = d;
#else
    // Fallback (should not trigger on gfx1250): plain fp32 FMA dot products.
    if (lane < 16) {
        const int c = c0 + lane;
        float s = 0.f;
#pragma unroll
        for (int f = 0; f < GAT_F; ++f)
            s = fmaf(W[f * GAT_NCELL + c], H[f * GAT_NCELL + c], s);
        out[c] = s;
    }
#endif
}

extern "C" void kernel_launch(void* const* d_in, const int* in_sizes, int n_in,
                              void* d_out, int out_size, void* d_ws, size_t ws_size,
                              hipStream_t stream) {
    (void)in_sizes; (void)n_in; (void)out_size; (void)d_ws; (void)ws_size;
    // setup_inputs order: 0=h, 1=adj, 2=W_pair, 3=a_pair, 4=W_out.
    // Only h and W_out influence the output (softmax columns sum to 1).
    const float* h     = (const float*)d_in[0];
    const float* w_out = (const float*)d_in[4];
    float*       out   = (float*)d_out;

    // 8 blocks x 256 threads = 64 wave32 waves x 16 cells = 1024 outputs.
    gat_collapsed_wmma<<<dim3(8), dim3(256), 0, stream>>>(h, w_out, out);
}